// FullAttention_64098091926048
// MI455X (gfx1250) — compile-verified
//
#include <hip/hip_runtime.h>
#include <hip/hip_bf16.h>

typedef __attribute__((ext_vector_type(2))) float v2f;
typedef __attribute__((ext_vector_type(8))) float v8f;
typedef unsigned int u32x4 __attribute__((ext_vector_type(4)));
typedef int i32x4 __attribute__((ext_vector_type(4)));
typedef int i32x8 __attribute__((ext_vector_type(8)));

#if __has_builtin(__builtin_amdgcn_tensor_load_to_lds)
#define HAVE_TDM 1
#if __has_include(<hip/amd_detail/amd_gfx1250_TDM.h>)
#define TDM_ARITY6 1
#endif
#endif

// ---- 16-lane-half reductions (wave32; xor of bits 0..3 never crosses halves)
__device__ __forceinline__ float half_max16(float v) {
#pragma unroll
  for (int m = 1; m < 16; m <<= 1) v = fmaxf(v, __shfl_xor(v, m, 32));
  return v;
}
__device__ __forceinline__ float half_sum16(float v) {
#pragma unroll
  for (int m = 1; m < 16; m <<= 1) v += __shfl_xor(v, m, 32);
  return v;
}

#define WMMA_F32(a, b, c) \
  __builtin_amdgcn_wmma_f32_16x16x4_f32(false, (a), false, (b), (short)0, (c), false, false)

// =====================================================================
// TDM: DMA a 2D fp32 tile [rows=256][cols=16], row stride 1024 elems,
// from global `gtile` into LDS at byte offset `lds_off` (row-major).
// D# layout per CDNA5 ISA ch.8 (group0 128b, group1 256b).
// Only called when HAVE_TDM is defined (device pass on gfx1250).
// =====================================================================
__device__ __forceinline__ void tdm_load_tile_256x16(unsigned int lds_off,
                                                     const float* gtile) {
#if defined(HAVE_TDM)
  const unsigned long long ga = (unsigned long long)(uintptr_t)gtile;
  u32x4 g0;
  g0[0] = 1u;                                   // count=1, user descriptor
  g0[1] = lds_off;                              // lds_addr (bytes)
  g0[2] = (unsigned int)(ga & 0xFFFFFFFFu);     // global_addr[31:0]
  g0[3] = (unsigned int)((ga >> 32) & 0x01FFFFFFu) | (2u << 30);  // ga[56:32] | type=2
  i32x8 g1;
  const unsigned int td0 = 1024u;               // tensor_dim0 (elems)
  const unsigned int td1 = 256u;                // tensor_dim1 (rows)
  g1[0] = (int)(2u << 16);                      // wg_mask=0, data_size=2 (4B)
  g1[1] = (int)((td0 & 0xFFFFu) << 16);         // tensor_dim0[15:0]
  g1[2] = (int)(((td0 >> 16) & 0xFFFFu) | ((td1 & 0xFFFFu) << 16));
  g1[3] = (int)(((td1 >> 16) & 0xFFFFu) | (16u << 16));  // tile_dim0 = 16
  g1[4] = (int)256u;                            // tile_dim1 = 256, tile_dim2 = 0
  g1[5] = (int)1024u;                           // tensor_dim0_stride[31:0]
  g1[6] = 0;                                    // stride0[47:32], stride1 lo
  g1[7] = 0;
  i32x4 z4 = {0, 0, 0, 0};
#if defined(TDM_ARITY6)
  i32x8 z8 = {0, 0, 0, 0, 0, 0, 0, 0};
  __builtin_amdgcn_tensor_load_to_lds(g0, g1, z4, z4, z8, 0);
#else
  __builtin_amdgcn_tensor_load_to_lds(g0, g1, z4, z4, 0);
#endif
#else
  (void)lds_off;
  (void)gtile;
#endif
}

__device__ __forceinline__ void wait_tensorcnt0() {
#if __has_builtin(__builtin_amdgcn_s_wait_tensorcnt)
  __builtin_amdgcn_s_wait_tensorcnt(0);
#else
  asm volatile("s_wait_tensorcnt 0x0" ::: "memory");
#endif
  asm volatile("" ::: "memory");  // keep LDS reads below the wait
}

// =====================================================================
// Kernel 1: QKV projection. qkv[b][o][n] = sum_c W[o][c]*x[b][c][n] + bias[o]
// Wave: 16 (o) x 64 (n) strip. grid = (nstrip=16, otile=48, b=16)
// =====================================================================
__global__ void qkv_kernel(const float* __restrict__ x, const float* __restrict__ w,
                           const float* __restrict__ bias, float* __restrict__ qkv) {
  const int lane = threadIdx.x;
  const int n0 = blockIdx.x * 64;
  const int o0 = blockIdx.y * 16;
  const int b  = blockIdx.z;
  const int half = lane >> 4, l16 = lane & 15, koff = half * 2;

  const float* xb = x + (size_t)b * 256 * 1024;
  const float* wrow = w + (size_t)(o0 + l16) * 256;

  v8f acc[4] = {};
  for (int k0 = 0; k0 < 256; k0 += 4) {
    v2f a;
    a.x = wrow[k0 + koff];
    a.y = wrow[k0 + koff + 1];
#pragma unroll
    for (int t = 0; t < 4; ++t) {
      const int n = n0 + t * 16 + l16;
      v2f bv;
      bv.x = xb[(size_t)(k0 + koff) * 1024 + n];
      bv.y = xb[(size_t)(k0 + koff + 1) * 1024 + n];
      acc[t] = WMMA_F32(a, bv, acc[t]);
    }
  }
  float* ob = qkv + (size_t)b * 768 * 1024;
#pragma unroll
  for (int r = 0; r < 8; ++r) {
    const int o = o0 + r + half * 8;
    const float bs = bias[o];
#pragma unroll
    for (int t = 0; t < 4; ++t)
      ob[(size_t)o * 1024 + n0 + t * 16 + l16] = acc[t][r] + bs;
  }
}

// =====================================================================
// Kernel 2: flash attention per 16-query tile, K/V/Q tiles staged in LDS
// by the Tensor Data Mover. grid = (qtile=64, b=16), block = 32 (1 wave)
// =====================================================================
__global__ void attn_kernel(const float* __restrict__ x, const float* __restrict__ qkv,
                            const float* __restrict__ alpha_p, float* __restrict__ x1) {
  __shared__ float ldsQ[256 * 16];  // [c][m]  16KB
  __shared__ float ldsK[256 * 16];  // [c][j]  16KB
  __shared__ float ldsV[256 * 16];  // [c][j]  16KB
  __shared__ float ldsP[16 * 16];   // P row-major, 1KB
  const int lane = threadIdx.x;
  const int qi0 = blockIdx.x * 16;
  const int b   = blockIdx.y;
  const int half = lane >> 4, l16 = lane & 15, koff = half * 2;
  const float alpha = alpha_p[0];
  const float scale = 0.0625f;  // 1/sqrt(256)

  const float* qg = qkv + (size_t)b * 768 * 1024;
  const float* kg = qg + 256 * 1024;
  const float* vg = qg + 512 * 1024;

#if defined(HAVE_TDM)
  const unsigned int offQ = (unsigned int)(uintptr_t)(void*)ldsQ;
  const unsigned int offK = (unsigned int)(uintptr_t)(void*)ldsK;
  const unsigned int offV = (unsigned int)(uintptr_t)(void*)ldsV;
  tdm_load_tile_256x16(offQ, qg + qi0);   // ldsQ[c][m] = q[c][qi0+m]
#else
  for (int i = lane; i < 256 * 16; i += 32)
    ldsQ[i] = qg[(size_t)(i >> 4) * 1024 + qi0 + (i & 15)];
#endif

  v8f O[16] = {};            // 16 c-tiles of [16q x 16c] accumulators
  float mrow[8], lrow[8];
#pragma unroll
  for (int r = 0; r < 8; ++r) { mrow[r] = -3.0e38f; lrow[r] = 0.0f; }

  for (int kt = 0; kt < 64; ++kt) {
    const int ki0 = kt * 16;
#if defined(HAVE_TDM)
    tdm_load_tile_256x16(offK, kg + ki0);  // ldsK[c][j] = k[c][ki0+j]
    tdm_load_tile_256x16(offV, vg + ki0);  // ldsV[c][j] = v[c][ki0+j]
    wait_tensorcnt0();
#else
    for (int i = lane; i < 256 * 16; i += 32) {
      ldsK[i] = kg[(size_t)(i >> 4) * 1024 + ki0 + (i & 15)];
      ldsV[i] = vg[(size_t)(i >> 4) * 1024 + ki0 + (i & 15)];
    }
    __syncthreads();
#endif
    // ---- S = Q^T K (K-loop over C=256), operands from LDS
    v8f S = {};
    for (int k0 = 0; k0 < 256; k0 += 4) {
      v2f a, bv;
      a.x = ldsQ[(k0 + koff) * 16 + l16];
      a.y = ldsQ[(k0 + koff + 1) * 16 + l16];
      bv.x = ldsK[(k0 + koff) * 16 + l16];
      bv.y = ldsK[(k0 + koff + 1) * 16 + l16];
      S = WMMA_F32(a, bv, S);
    }
    // ---- online softmax
#pragma unroll
    for (int r = 0; r < 8; ++r) {
      float p = S[r] * scale;
      const float tm = half_max16(p);
      const float nm = fmaxf(mrow[r], tm);
      const float corr = __expf(mrow[r] - nm);
      mrow[r] = nm;
      p = __expf(p - nm);
      lrow[r] = lrow[r] * corr + half_sum16(p);
#pragma unroll
      for (int t = 0; t < 16; ++t) O[t][r] *= corr;
      ldsP[(r + half * 8) * 16 + l16] = p;
    }
    __syncthreads();
    // ---- reload P as A fragments (4 K-chunks of 4)
    v2f pa[4];
#pragma unroll
    for (int kc = 0; kc < 4; ++kc) {
      pa[kc].x = ldsP[l16 * 16 + kc * 4 + koff];
      pa[kc].y = ldsP[l16 * 16 + kc * 4 + koff + 1];
    }
    // ---- O += P * V^T ; B[k][c] = ldsV[c][k], contiguous v2f LDS reads
#pragma unroll
    for (int t = 0; t < 16; ++t) {
      const float* vrow = &ldsV[(t * 16 + l16) * 16];
#pragma unroll
      for (int kc = 0; kc < 4; ++kc) {
        v2f bv = *(const v2f*)(vrow + kc * 4 + koff);
        O[t] = WMMA_F32(pa[kc], bv, O[t]);
      }
    }
    __syncthreads();
  }
  // ---- epilogue: normalize + fused residual relu at flat index qi*256+c
  const float* xb = x + (size_t)b * 262144;
  float* x1b = x1 + (size_t)b * 262144;
#pragma unroll
  for (int r = 0; r < 8; ++r) {
    const float inv = 1.0f / lrow[r];
    const int qi = qi0 + r + half * 8;
#pragma unroll
    for (int t = 0; t < 16; ++t) {
      const size_t idx = (size_t)qi * 256 + t * 16 + l16;
      x1b[idx] = xb[idx] + alpha * fmaxf(O[t][r] * inv, 0.0f);
    }
  }
}

// =====================================================================
// Kernel 3: 3x3 SAME conv as implicit GEMM + bias + residual relu.
// grid = (nstrip=16, otile=16, b=16), block = 32
// =====================================================================
__global__ void conv_kernel(const float* __restrict__ x1, const float* __restrict__ w,
                            const float* __restrict__ bias, const float* __restrict__ alpha_p,
                            float* __restrict__ out) {
  const int lane = threadIdx.x;
  const int n0 = blockIdx.x * 64;
  const int o0 = blockIdx.y * 16;
  const int b  = blockIdx.z;
  const int half = lane >> 4, l16 = lane & 15, koff = half * 2;
  const float alpha = alpha_p[0];
  const float* xb = x1 + (size_t)b * 262144;
  const float* wrow = w + (size_t)(o0 + l16) * 2304;

  v8f acc[4] = {};
  for (int kh = 0; kh < 3; ++kh) {
    for (int kw = 0; kw < 3; ++kw) {
      for (int c0 = 0; c0 < 256; c0 += 4) {
        v2f a;
        a.x = wrow[(c0 + koff) * 9 + kh * 3 + kw];
        a.y = wrow[(c0 + koff + 1) * 9 + kh * 3 + kw];
#pragma unroll
        for (int t = 0; t < 4; ++t) {
          const int n = n0 + t * 16;
          const int hh = (n >> 5) + kh - 1;
          const int ww = (n & 31) + l16 + kw - 1;
          const bool ok = (hh >= 0) && (hh < 32) && (ww >= 0) && (ww < 32);
          const float* bp = xb + (size_t)(c0 + koff) * 1024 + hh * 32 + ww;
          v2f bv;
          bv.x = ok ? bp[0] : 0.0f;
          bv.y = ok ? bp[1024] : 0.0f;
          acc[t] = WMMA_F32(a, bv, acc[t]);
        }
      }
    }
  }
  float* ob = out + (size_t)b * 262144;
#pragma unroll
  for (int r = 0; r < 8; ++r) {
    const int o = o0 + r + half * 8;
    const float bs = bias[o];
#pragma unroll
    for (int t = 0; t < 4; ++t) {
      const size_t idx = (size_t)o * 1024 + n0 + t * 16 + l16;
      ob[idx] = xb[idx] + alpha * fmaxf(acc[t][r] + bs, 0.0f);
    }
  }
}

extern "C" void kernel_launch(void* const* d_in, const int* in_sizes, int n_in,
                              void* d_out, int out_size, void* d_ws, size_t ws_size,
                              hipStream_t stream) {
  const float* x      = (const float*)d_in[0];
  const float* w_qkv  = (const float*)d_in[1];
  const float* b_qkv  = (const float*)d_in[2];
  const float* w_out  = (const float*)d_in[3];
  const float* b_out  = (const float*)d_in[4];
  const float* alpha  = (const float*)d_in[5];
  float* out = (float*)d_out;

  float* qkv = (float*)d_ws;                       // 48MB
  float* x1  = qkv + (size_t)16 * 768 * 1024;      // 16MB

  qkv_kernel<<<dim3(16, 48, 16), 32, 0, stream>>>(x, w_qkv, b_qkv, qkv);
  attn_kernel<<<dim3(64, 16), 32, 0, stream>>>(x, qkv, alpha, x1);
  conv_kernel<<<dim3(16, 16, 16), 32, 0, stream>>>(x1, w_out, b_out, alpha, out);
}